// NeuralDisCoCirc_39779987095596
// MI455X (gfx1250) — compile-verified
//
#include <hip/hip_runtime.h>

// Problem constants (reference: B=32, L=64, D=512, fp32 everywhere)
#define B_  32
#define L_  64
#define D_  512

#define BLOCK_THREADS    128                 // 4 waves (wave32)
#define BLOCKS_PER_BATCH 8                   // 8 blocks * 4 waves = 32 waves/batch
#define WAVES_PER_BATCH  32                  // each wave owns 16 output columns

typedef float v2f __attribute__((ext_vector_type(2)));
typedef float v8f __attribute__((ext_vector_type(8)));

// Zero the per-(batch,layer) barrier counters each call (ws is not re-poisoned
// between graph replays, so we must reset deterministically every launch).
__global__ void ndc_zero_cnt(unsigned* __restrict__ cnt, int n) {
    int i = blockIdx.x * blockDim.x + threadIdx.x;
    if (i < n) cnt[i] = 0u;
}

__global__ __launch_bounds__(BLOCK_THREADS)
void ndc_chain_kernel(const float* __restrict__ x_in,     // [B, D]
                      const float* __restrict__ W,        // [B, L, D, D]  (i-major, j contiguous)
                      const float* __restrict__ bias,     // [B, L, D]
                      float*       __restrict__ out,      // [B, D]
                      float*       __restrict__ xbuf,     // [2, B, D] double buffer (ws)
                      unsigned*    __restrict__ cnt)      // [B, L] arrive counters (ws)
{
    const int lane        = threadIdx.x & 31;
    const int waveInBlock = threadIdx.x >> 5;
    const int batch       = blockIdx.x / BLOCKS_PER_BATCH;
    const int wgInBatch   = blockIdx.x % BLOCKS_PER_BATCH;
    const int waveInBatch = wgInBatch * (BLOCK_THREADS / 32) + waveInBlock;
    const int j0          = waveInBatch * 16;      // this wave's 16 output columns

    // A/B/C operand lane decomposition for 16x16x4 f32 WMMA:
    //   lanes 0-15 carry K={0,1}, lanes 16-31 carry K={2,3}
    const int m   = lane & 15;        // M index within tile (output column offset)
    const int hi  = lane >> 4;        // 0 or 1
    const int klo = hi << 1;          // K base carried by this half-wave (0 or 2)

    for (int l = 0; l < L_; ++l) {
        const float* xcur = (l == 0)
            ? (x_in + (size_t)batch * D_)
            : (xbuf + ((size_t)(l & 1) * B_ + batch) * D_);
        const float* Wl = W    + (((size_t)batch * L_ + l) * D_) * D_;
        const float* bl = bias + ((size_t)batch * L_ + l) * D_;

        // Seed accumulator with bias, replicated across the (redundant) N dim.
        // C/D layout: VGPR r at lane t holds element (M = r + 8*(t>=16), N = t&15).
        v8f c;
        {
            const float4 b0 = *(const float4*)(bl + j0 + 8 * hi);
            const float4 b1 = *(const float4*)(bl + j0 + 8 * hi + 4);
            c[0] = b0.x; c[1] = b0.y; c[2] = b0.z; c[3] = b0.w;
            c[4] = b1.x; c[5] = b1.y; c[6] = b1.z; c[7] = b1.w;
        }

        // A-tile column base for this lane: W rows advance by D_ floats.
        const float* Acol = Wl + j0 + m;

        // out[j0+m] = bias + sum_i x[i] * W[i][j0+m]; K blocked by 4 per WMMA.
        // Weights are streamed once -> non-temporal loads (don't pollute L2).
        // unroll 16 => ~4 KB of weight loads in flight per wave for latency cover.
        #pragma unroll 16
        for (int i0 = 0; i0 < D_; i0 += 4) {
            v2f a, b;
            // A (16x4 f32): VGPR0 = (M=m, K=klo), VGPR1 = (M=m, K=klo+1)
            a.x = __builtin_nontemporal_load(Acol + (size_t)(i0 + klo)     * D_);
            a.y = __builtin_nontemporal_load(Acol + (size_t)(i0 + klo + 1) * D_);
            // B (4x16 f32): row K broadcast across all 16 N columns
            b = *(const v2f*)(xcur + i0 + klo);
            c = __builtin_amdgcn_wmma_f32_16x16x4_f32(
                    /*neg_a=*/false, a, /*neg_b=*/false, b,
                    /*c_mod=*/(short)0, c, /*reuse_a=*/false, /*reuse_b=*/false);
        }

        // All 16 N columns of D are identical; lanes with N==0 (lane 0 and 16)
        // each hold 8 of the 16 results -> two float4 stores.
        float* dst = (l == L_ - 1)
            ? (out  + (size_t)batch * D_)
            : (xbuf + ((size_t)((l + 1) & 1) * B_ + batch) * D_);
        if (m == 0) {
            *(float4*)(dst + j0 + 8 * hi)     = make_float4(c[0], c[1], c[2], c[3]);
            *(float4*)(dst + j0 + 8 * hi + 4) = make_float4(c[4], c[5], c[6], c[7]);
        }

        // Per-batch cross-workgroup barrier: all 32 waves of this batch must
        // publish their x-slice before anyone starts layer l+1.
        if (l != L_ - 1) {
            unsigned* ctr = cnt + batch * L_ + l;
            if (lane == 0) {
                __hip_atomic_fetch_add(ctr, 1u, __ATOMIC_RELEASE,
                                       __HIP_MEMORY_SCOPE_AGENT);
            }
            // Warm the next layer's weight tile while we wait.
            const float* Wn = W + (((size_t)batch * L_ + (l + 1)) * D_) * D_ + j0 + m;
            __builtin_prefetch(Wn, 0, 1);
            __builtin_prefetch(Wn + D_, 0, 1);
            while (__hip_atomic_load(ctr, __ATOMIC_ACQUIRE,
                                     __HIP_MEMORY_SCOPE_AGENT) < WAVES_PER_BATCH) {
                __builtin_amdgcn_s_sleep(1);
            }
        }
    }
}

extern "C" void kernel_launch(void* const* d_in, const int* in_sizes, int n_in,
                              void* d_out, int out_size, void* d_ws, size_t ws_size,
                              hipStream_t stream) {
    const float* x    = (const float*)d_in[0];   // [B, D]
    const float* W    = (const float*)d_in[1];   // [B, L, D, D]
    const float* bias = (const float*)d_in[2];   // [B, L, D]
    float* out = (float*)d_out;                  // [B, D]

    // Workspace layout: xbuf (2*B*D floats = 128 KB) | counters (B*L u32 = 8 KB)
    float*    xbuf = (float*)d_ws;
    unsigned* cnt  = (unsigned*)((char*)d_ws + (size_t)2 * B_ * D_ * sizeof(float));

    ndc_zero_cnt<<<(B_ * L_ + 255) / 256, 256, 0, stream>>>(cnt, B_ * L_);
    ndc_chain_kernel<<<B_ * BLOCKS_PER_BATCH, BLOCK_THREADS, 0, stream>>>(
        x, W, bias, out, xbuf, cnt);
}